// FastSSMLayer_24240795419335
// MI455X (gfx1250) — compile-verified
//
#include <hip/hip_runtime.h>

typedef float v2f __attribute__((ext_vector_type(2)));
typedef float v8f __attribute__((ext_vector_type(8)));

#define D_MODEL 1024
#define D_STATE 64
#define BATCH   4
#define SEQ     4096
#define LCHUNK  64
#define NCHUNK  (SEQ / LCHUNK)   // 64 chunks per sequence

// ---------------------------------------------------------------------------
// Phase 1: Bu = x @ B^T for one (batch, chunk) tile (64 timesteps x 64 states)
// via V_WMMA_F32_16X16X4_F32, then local (zero-init) scan inside the chunk.
// Writes local-h tile to hloc and the chunk-end state to ends.
// x loads use default RT temporal hint so x stays resident in the 192MB L2
// for the phase-3 re-read (x is 64MB total).
// ---------------------------------------------------------------------------
__global__ __launch_bounds__(256) void ssm_phase1(
    const float* __restrict__ x, const float* __restrict__ Amat,
    const float* __restrict__ Bmat, float* __restrict__ hloc,
    float* __restrict__ ends)
{
  __shared__ float tile[LCHUNK][D_STATE];   // 16 KB

  const int wg    = blockIdx.x;
  const int b     = wg >> 6;
  const int chunk = wg & 63;
  const int l0    = chunk * LCHUNK;
  const int tid   = threadIdx.x;
  const int w     = tid >> 5;               // wave id 0..7
  const int lane  = tid & 31;

  // wave w owns row-block mi (16 timesteps) and two 16-state column blocks
  const int mi = w >> 1;
  const int nb = (w & 1) * 2;

  const int m  = lane & 15;                 // A row / B column within block
  const int kb = (lane >> 4) * 2;           // K sub-offset per half-wave

  const float* xrow = x    + ((size_t)b * SEQ + l0 + mi * 16 + m) * D_MODEL + kb;
  const float* br0  = Bmat + (size_t)((nb + 0) * 16 + m) * D_MODEL + kb;
  const float* br1  = Bmat + (size_t)((nb + 1) * 16 + m) * D_MODEL + kb;

  v8f acc0 = {};
  v8f acc1 = {};
#pragma unroll 8
  for (int k = 0; k < D_MODEL; k += 4) {
    v2f av = *(const v2f*)(xrow + k);
    v2f b0 = *(const v2f*)(br0 + k);
    v2f b1 = *(const v2f*)(br1 + k);
    acc0 = __builtin_amdgcn_wmma_f32_16x16x4_f32(false, av, false, b0,
                                                 (short)0, acc0, false, false);
    acc1 = __builtin_amdgcn_wmma_f32_16x16x4_f32(false, av, false, b1,
                                                 (short)0, acc1, false, false);
  }

  // Spill Bu tile to LDS (C/D layout: VGPR v -> row v + 8*(lane>=16))
  const int rowoff = mi * 16 + (lane >> 4) * 8;
#pragma unroll
  for (int v = 0; v < 8; ++v) {
    tile[rowoff + v][(nb + 0) * 16 + m] = acc0[v];
    tile[rowoff + v][(nb + 1) * 16 + m] = acc1[v];
  }
  __syncthreads();

  // Local scan, h_{-1} = 0:  h_l = a*h_{l-1} + Bu_l
  if (tid < D_STATE) {
    const int   s = tid;
    const float a = Amat[s * (D_STATE + 1)];   // diag(A)
    float h = 0.f;
    for (int l = 0; l < LCHUNK; ++l) {
      h = fmaf(a, h, tile[l][s]);
      tile[l][s] = h;
    }
    ends[((size_t)b * NCHUNK + chunk) * D_STATE + s] = h;
  }
  __syncthreads();

  // Bulk-copy local-h tile to global workspace (float4, coalesced).
  // hloc is re-read in phase 3: keep RT so it lands in L2.
  float4*       dst = (float4*)(hloc + ((size_t)b * SEQ + l0) * D_STATE);
  const float4* src = (const float4*)(&tile[0][0]);
#pragma unroll
  for (int r = 0; r < 4; ++r) dst[r * 256 + tid] = src[r * 256 + tid];
}

// ---------------------------------------------------------------------------
// Phase 2: chunk-prefix carries. carry_in(chunk c) for each (b,s) sequence:
//   carry_{c} = a^64 * carry_{c-1} + end_{c-1}
// 256 threads = one per (b,s); loads all ends first, register-only chain.
// ---------------------------------------------------------------------------
__global__ __launch_bounds__(256) void ssm_phase2(
    const float* __restrict__ Amat, const float* __restrict__ ends,
    float* __restrict__ carries)
{
  const int tid = threadIdx.x;
  const int b   = tid >> 6;
  const int s   = tid & 63;

  const float a = Amat[s * (D_STATE + 1)];
  float a64 = a;
#pragma unroll
  for (int i = 0; i < 6; ++i) a64 *= a64;    // a^64

  float e[NCHUNK];
#pragma unroll
  for (int c = 0; c < NCHUNK; ++c)
    e[c] = ends[((size_t)b * NCHUNK + c) * D_STATE + s];

  float carry = 0.f;
#pragma unroll
  for (int c = 0; c < NCHUNK; ++c) {
    carries[((size_t)b * NCHUNK + c) * D_STATE + s] = carry;
    carry = fmaf(a64, carry, e[c]);
  }
}

// ---------------------------------------------------------------------------
// Phase 3: fix up h with incoming carry (h_l += a^{l+1} * carry), then
// y = h @ C^T + D*x via WMMA, fused epilogue writing y directly.
// x re-reads are last-use -> non-temporal loads (don't re-retain in L2);
// y is write-once/never-read -> non-temporal stores (don't evict x/hloc).
// ---------------------------------------------------------------------------
__global__ __launch_bounds__(256) void ssm_phase3(
    const float* __restrict__ x, const float* __restrict__ Amat,
    const float* __restrict__ Cmat, const float* __restrict__ Dvec,
    const float* __restrict__ hloc, const float* __restrict__ carries,
    float* __restrict__ y)
{
  __shared__ float tile[LCHUNK][D_STATE];   // 16 KB

  const int wg    = blockIdx.x;
  const int b     = wg >> 6;
  const int chunk = wg & 63;
  const int l0    = chunk * LCHUNK;
  const int tid   = threadIdx.x;
  const int w     = tid >> 5;
  const int lane  = tid & 31;

  // Load local-h tile into LDS
  const float4* src = (const float4*)(hloc + ((size_t)b * SEQ + l0) * D_STATE);
  float4*       dst = (float4*)(&tile[0][0]);
#pragma unroll
  for (int r = 0; r < 4; ++r) dst[r * 256 + tid] = src[r * 256 + tid];
  __syncthreads();

  // Carry fix-up: h_l += a^{l+1} * carry_in
  if (tid < D_STATE) {
    const int   s = tid;
    const float a = Amat[s * (D_STATE + 1)];
    float val = carries[((size_t)b * NCHUNK + chunk) * D_STATE + s];
    for (int l = 0; l < LCHUNK; ++l) {
      val *= a;
      tile[l][s] += val;
    }
  }
  __syncthreads();

  // GEMM: M=64 (timesteps), K=64 (states), N=1024 (d_model)
  const int mi    = w >> 1;          // row block (16 timesteps)
  const int nbase = (w & 1) * 32;    // 32 N sub-tiles per wave
  const int m     = lane & 15;
  const int kb    = (lane >> 4) * 2;

  // Hoist A fragments for this row block (depend only on mi,k)
  v2f afrag[16];
#pragma unroll
  for (int kk = 0; kk < 16; ++kk)
    afrag[kk] = *(const v2f*)(&tile[mi * 16 + m][kk * 4 + kb]);

  const int rowoff = (lane >> 4) * 8;
#pragma unroll 1
  for (int nt = 0; nt < 32; ++nt) {
    const int    d    = (nbase + nt) * 16 + m;            // output column
    const float* crow = Cmat + (size_t)d * D_STATE + kb;  // C is (1024,64)

    v8f acc = {};
#pragma unroll
    for (int kk = 0; kk < 16; ++kk) {
      v2f bf = *(const v2f*)(crow + kk * 4);
      acc = __builtin_amdgcn_wmma_f32_16x16x4_f32(false, afrag[kk], false, bf,
                                                  (short)0, acc, false, false);
    }

    const float  Dd   = Dvec[d];
    const size_t base = ((size_t)b * SEQ + l0 + mi * 16 + rowoff) * D_MODEL + d;
#pragma unroll
    for (int v = 0; v < 8; ++v) {
      const size_t off = base + (size_t)v * D_MODEL;
      const float  xv  = __builtin_nontemporal_load(x + off);   // last use of x
      __builtin_nontemporal_store(fmaf(Dd, xv, acc[v]), y + off);
    }
  }
}

// ---------------------------------------------------------------------------
extern "C" void kernel_launch(void* const* d_in, const int* in_sizes, int n_in,
                              void* d_out, int out_size, void* d_ws, size_t ws_size,
                              hipStream_t stream) {
  const float* x = (const float*)d_in[0];
  const float* A = (const float*)d_in[1];
  const float* B = (const float*)d_in[2];
  const float* C = (const float*)d_in[3];
  const float* D = (const float*)d_in[4];
  float* y = (float*)d_out;

  char*  ws      = (char*)d_ws;
  float* hloc    = (float*)ws;                                        // 4 MB
  float* ends    = (float*)(ws + (size_t)BATCH * SEQ * D_STATE * 4);  // 64 KB
  float* carries = ends + BATCH * NCHUNK * D_STATE;                   // 64 KB

  dim3 blk(256);
  dim3 grid(BATCH * NCHUNK);   // 256 workgroups: one per (batch, chunk)

  ssm_phase1<<<grid, blk, 0, stream>>>(x, A, B, hloc, ends);
  ssm_phase2<<<1,    blk, 0, stream>>>(A, ends, carries);
  ssm_phase3<<<grid, blk, 0, stream>>>(x, A, C, D, hloc, carries, y);
}